// SparseNetwork_16801912062197
// MI455X (gfx1250) — compile-verified
//
#include <hip/hip_runtime.h>

typedef float v2f __attribute__((ext_vector_type(2)));
typedef float v8f __attribute__((ext_vector_type(8)));

// 6 weight sets: FW (4096 nets), HW0..HW3 (4096 each), LW (2048). Total 22528 nets.
// ws coefficient slots: A[0:64) H0[64:128) H1[128:192) H2[192:256) H3[256:320) L[320:352)
struct Ptrs {
    const float* W1[6];
    const float* W2[6];
    const float* W3[6];
};

__device__ __forceinline__ void net_decode(int gid, int& set, int& local) {
    if (gid < 4096)       { set = 0; local = gid; }
    else if (gid < 20480) { set = 1 + ((gid - 4096) >> 12); local = (gid - 4096) & 4095; }
    else                  { set = 5; local = gid - 20480; }
}

__device__ __forceinline__ int ws_index(int gid) {
    int set, local;
    net_decode(gid, set, local);
    int base = (set < 5) ? set * 64 : 320;
    return base + (local >> 6);   // groups of 64 consecutive nets
}

// Kernel 1: one net per thread; c_n = W3 . (W2 . W1[:,4]); hierarchical reduce:
// LDS ds_add_f32 within the block (each block covers exactly 4 groups of 64),
// then 4 global atomics per block into the 352-slot workspace.
__global__ __launch_bounds__(256) void coeff_kernel(Ptrs p, float* __restrict__ ws) {
    __shared__ float part[4];
    int tid = threadIdx.x;
    if (tid < 4) part[tid] = 0.0f;
    __syncthreads();

    int gid = blockIdx.x * 256 + tid;           // 0..22527 (88 blocks * 256)
    int set, local;
    net_decode(gid, set, local);
    const float* w1 = p.W1[set] + local * 10;   // [n,2,5]
    const float* w2 = p.W2[set] + local * 4;    // [n,2,2]
    const float* w3 = p.W3[set] + local * 2;    // [n,1,2]
    float a0 = w1[4];                           // W1[n,0,4]
    float a1 = w1[9];                           // W1[n,1,4]
    float m0 = w2[0] * a0 + w2[1] * a1;
    float m1 = w2[2] * a0 + w2[3] * a1;
    float c  = w3[0] * m0 + w3[1] * m1;

    atomicAdd(&part[tid >> 6], c);              // ds_add_f32
    __syncthreads();

    if (tid < 4) {
        int rep = blockIdx.x * 256 + tid * 64;  // first net of this group
        atomicAdd(&ws[ws_index(rep)], part[tid]);
    }
}

// Kernel 2: scalar residual chain -> F[32], then out[64x32] = x[64x64] @ G,
// G[d,o] = F[o], computed with V_WMMA_F32_16X16X4_F32 (8 waves, one 16x16 tile each,
// 16 K-steps of 4).
__global__ __launch_bounds__(512) void chain_gemm_kernel(const float* __restrict__ x,
                                                         const float* __restrict__ ws,
                                                         float* __restrict__ out) {
    __shared__ float coeff[352];
    __shared__ float F[32];
    __shared__ float g[64];
    __shared__ float r[64];

    int tid = threadIdx.x;
    if (tid < 352) coeff[tid] = ws[tid];
    __syncthreads();

    if (tid == 0) {
        float s;
        for (int v = 0; v < 64; ++v) g[v] = coeff[v];                  // t0 coeffs (A)
        // i=0 (even): residual = g; g = H0 * sum(g)
        s = 0.0f; for (int v = 0; v < 64; ++v) { r[v] = g[v]; s += g[v]; }
        for (int v = 0; v < 64; ++v) g[v] = coeff[64 + v] * s;
        // i=1 (odd): g = H1 * sum(g) + residual
        s = 0.0f; for (int v = 0; v < 64; ++v) s += g[v];
        for (int v = 0; v < 64; ++v) g[v] = coeff[128 + v] * s + r[v];
        // i=2 (even): residual = g; g = H2 * sum(g)
        s = 0.0f; for (int v = 0; v < 64; ++v) { r[v] = g[v]; s += g[v]; }
        for (int v = 0; v < 64; ++v) g[v] = coeff[192 + v] * s;
        // i=3 (odd): g = H3 * sum(g) + residual
        s = 0.0f; for (int v = 0; v < 64; ++v) s += g[v];
        for (int v = 0; v < 64; ++v) g[v] = coeff[256 + v] * s + r[v];
        // last layer: F[o] = L[o] * sum(g)
        s = 0.0f; for (int v = 0; v < 64; ++v) s += g[v];
        for (int o = 0; o < 32; ++o) F[o] = coeff[320 + o] * s;
    }
    __syncthreads();

    int wid  = tid >> 5;
    int lane = tid & 31;
    if (wid < 8) {                     // wave-uniform branch: EXEC all-ones inside
        int mi = wid >> 1, ni = wid & 1;
        int m_base = mi * 16, n_base = ni * 16;

        // B tile (4x16): G is constant along K -> both VGPRs hold F[n]
        float fb = F[n_base + (lane & 15)];
        v2f b; b.x = fb; b.y = fb;

        v8f c = {};
        // A tile layout (16x4 f32): lanes 0-15 -> K=k+0,k+1 ; lanes 16-31 -> K=k+2,k+3
        int row  = m_base + (lane & 15);
        int ksel = (lane >> 4) << 1;
        #pragma unroll
        for (int k = 0; k < 64; k += 4) {
            v2f a;
            a.x = x[row * 64 + k + ksel];
            a.y = x[row * 64 + k + ksel + 1];
            c = __builtin_amdgcn_wmma_f32_16x16x4_f32(
                    /*neg_a=*/false, a, /*neg_b=*/false, b,
                    /*c_mod=*/(short)0, c, /*reuse_a=*/false, /*reuse_b=*/false);
        }

        // D layout: VGPR rr, lanes 0-15: M=m_base+rr ; lanes 16-31: M=m_base+8+rr
        int col    = n_base + (lane & 15);
        int rowAdd = (lane >> 4) * 8;
        #pragma unroll
        for (int rr = 0; rr < 8; ++rr) {
            out[(m_base + rowAdd + rr) * 32 + col] = c[rr];
        }
    }
}

extern "C" void kernel_launch(void* const* d_in, const int* in_sizes, int n_in,
                              void* d_out, int out_size, void* d_ws, size_t ws_size,
                              hipStream_t stream) {
    (void)in_sizes; (void)n_in; (void)out_size; (void)ws_size;
    // dict order: x, FW1, FW2, FW3, HW1[0..3], HW2[0..3], HW3[0..3], LW1, LW2, LW3
    const float* x = (const float*)d_in[0];
    Ptrs p;
    p.W1[0] = (const float*)d_in[1];
    p.W2[0] = (const float*)d_in[2];
    p.W3[0] = (const float*)d_in[3];
    for (int i = 0; i < 4; ++i) {
        p.W1[1 + i] = (const float*)d_in[4 + i];
        p.W2[1 + i] = (const float*)d_in[8 + i];
        p.W3[1 + i] = (const float*)d_in[12 + i];
    }
    p.W1[5] = (const float*)d_in[16];
    p.W2[5] = (const float*)d_in[17];
    p.W3[5] = (const float*)d_in[18];

    float* ws = (float*)d_ws;
    hipMemsetAsync(ws, 0, 352 * sizeof(float), stream);
    coeff_kernel<<<88, 256, 0, stream>>>(p, ws);
    chain_gemm_kernel<<<1, 512, 0, stream>>>(x, ws, (float*)d_out);
}